// KGAggregator_20864951124276
// MI455X (gfx1250) — compile-verified
//
#include <hip/hip_runtime.h>
#include <math.h>

typedef __attribute__((ext_vector_type(2))) float v2f;
typedef __attribute__((ext_vector_type(8))) float v8f;

#define FMIN 1e-15f
#define MAXN 0.996f   /* (1 - PROJ_EPS)/sqrt(c), c=1 */

__device__ __forceinline__ float artanhf_(float x) {
    x = fminf(fmaxf(x, -1.0f + 1e-7f), 1.0f - 1e-7f);
    return 0.5f * logf((1.0f + x) / (1.0f - x));
}

// ---------------------------------------------------------------------------
// Kernel A: per-b hyperbolic KG + user aggregation.  One block (256 thr) per b.
// Streams the 256KB Wr[b] slice (the bandwidth-dominant term) with float4
// loads; everything else lives in LDS.  Writes out1[b][64] (pre-hyp_linear).
// ---------------------------------------------------------------------------
__global__ void __launch_bounds__(256) hyp_agg_kernel(
    const float* __restrict__ self_v,   // [B,1,64]
    const float* __restrict__ ungh_v,   // [B,16,64]
    const float* __restrict__ engh_v,   // [B,1,16,64]
    const float* __restrict__ item_v,   // [B,64]
    const float* __restrict__ Wr,       // [B,1,16,64,64]
    const float* __restrict__ Wui,      // [64,64]
    float* __restrict__ out1)           // [B,64]
{
    const int b = blockIdx.x;
    const int t = threadIdx.x;

    __shared__ float s_self[64], s_item[64], s_tself[64], s_titem[64];
    __shared__ float s_engh[1024], s_ungh[1024];
    __shared__ float s_te[1024];     // t_engh, later sub_e
    __shared__ float s_tu[1024];     // t_ungh, later sub_u
    __shared__ float s_tmp[1024];    // per-(s,r) einsum products
    __shared__ float s_Wui[4096];
    __shared__ float s_part[256];
    __shared__ float s_e[16], s_att[16];
    __shared__ float s_ns2e[16], s_dote[16], s_ns2u[16], s_dotu[16];
    __shared__ float s_A[16], s_den[16], s_k[16], s_coef[16];
    __shared__ float s_agg[64], s_kg[64], s_ua[64], s_m1[64], s_imW[64];
    __shared__ float s_scal[16];

    // ---- load ----
    if (t < 64) { s_self[t] = self_v[b * 64 + t]; s_item[t] = item_v[b * 64 + t]; }
    for (int i = t; i < 1024; i += 256) {
        s_engh[i] = engh_v[(size_t)b * 1024 + i];
        s_ungh[i] = ungh_v[(size_t)b * 1024 + i];
    }
    for (int i = t; i < 4096; i += 256) s_Wui[i] = Wui[i];
    __syncthreads();

    // ---- |self|^2, |item|^2 ----
    s_part[t] = 0.f;
    if (t < 64)       s_part[t] = s_self[t] * s_self[t];
    else if (t < 128) s_part[t] = s_item[t - 64] * s_item[t - 64];
    __syncthreads();
    if (t == 0) {
        float a = 0.f, c2 = 0.f;
        for (int i = 0; i < 64; ++i) { a += s_part[i]; c2 += s_part[64 + i]; }
        s_scal[0] = a; s_scal[1] = c2;
    }
    __syncthreads();
    const float x2s = s_scal[0], x2i = s_scal[1];

    // ---- logmap0(self), logmap0(item) ----
    {
        float ns = fmaxf(sqrtf(x2s), FMIN), cs = artanhf_(ns) / ns;
        float ni = fmaxf(sqrtf(x2i), FMIN), ci = artanhf_(ni) / ni;
        if (t < 64) { s_tself[t] = cs * s_self[t]; s_titem[t] = ci * s_item[t]; }
    }
    __syncthreads();

    // ---- row sums: |engh_s|^2, |ungh_s|^2, dot(self,engh_s), dot(item,ungh_s)
    {
        int s = t >> 4, c0 = (t & 15) * 4, base = s * 64 + c0;
        float pe = 0, pu = 0, de = 0, du = 0;
        for (int q = 0; q < 4; ++q) {
            float ev = s_engh[base + q], uv = s_ungh[base + q];
            pe += ev * ev;           pu += uv * uv;
            de += s_self[c0 + q] * ev; du += s_item[c0 + q] * uv;
        }
        s_part[t] = pe; __syncthreads();
        if (t < 16) { float a = 0; for (int i = 0; i < 16; ++i) a += s_part[t * 16 + i]; s_ns2e[t] = a; }
        __syncthreads();
        s_part[t] = pu; __syncthreads();
        if (t < 16) { float a = 0; for (int i = 0; i < 16; ++i) a += s_part[t * 16 + i]; s_ns2u[t] = a; }
        __syncthreads();
        s_part[t] = de; __syncthreads();
        if (t < 16) { float a = 0; for (int i = 0; i < 16; ++i) a += s_part[t * 16 + i]; s_dote[t] = a; }
        __syncthreads();
        s_part[t] = du; __syncthreads();
        if (t < 16) { float a = 0; for (int i = 0; i < 16; ++i) a += s_part[t * 16 + i]; s_dotu[t] = a; }
        __syncthreads();
    }

    // ---- t_engh ----
    if (t < 16) { float n = fmaxf(sqrtf(s_ns2e[t]), FMIN); s_coef[t] = artanhf_(n) / n; }
    __syncthreads();
    for (int i = t; i < 1024; i += 256) s_te[i] = s_coef[i >> 6] * s_engh[i];
    __syncthreads();
    // ---- t_ungh ----
    if (t < 16) { float n = fmaxf(sqrtf(s_ns2u[t]), FMIN); s_coef[t] = artanhf_(n) / n; }
    __syncthreads();
    for (int i = t; i < 1024; i += 256) s_tu[i] = s_coef[i >> 6] * s_ungh[i];
    __syncthreads();

    // ---- the heavy einsum: e[s] = sum_r (sum_c t_self[c]*Wr[b,s,r,c]) * t_engh[s,r]
    {
        const size_t wrbase = (size_t)b * 16 * 64 * 64;
        for (int k = 0; k < 4; ++k) {
            int idx = t + k * 256;
            int s = idx >> 6, r = idx & 63;
            const float4* w4 = (const float4*)(Wr + wrbase + (size_t)s * 4096 + (size_t)r * 64);
            float acc = 0.f;
#pragma unroll
            for (int q = 0; q < 16; ++q) {
                float4 w = w4[q];
                acc += w.x * s_tself[4 * q]     + w.y * s_tself[4 * q + 1]
                     + w.z * s_tself[4 * q + 2] + w.w * s_tself[4 * q + 3];
            }
            s_tmp[idx] = acc * s_te[idx];
        }
    }
    __syncthreads();
    if (t < 16) { float a = 0; for (int r = 0; r < 64; ++r) a += s_tmp[t * 64 + r]; s_e[t] = a; }
    __syncthreads();
    if (t == 0) {   // softmax over s
        float m = -1e30f; for (int s = 0; s < 16; ++s) m = fmaxf(m, s_e[s]);
        float sum = 0.f, ex[16];
        for (int s = 0; s < 16; ++s) { ex[s] = expf(s_e[s] - m); sum += ex[s]; }
        for (int s = 0; s < 16; ++s) s_att[s] = ex[s] / sum;
    }
    __syncthreads();

    // ---- ngh_log[s] = logmap(self, engh_s);  agg = sum_s att*ngh_log ----
    if (t < 16) {
        float xy = -s_dote[t], y2 = s_ns2e[t];
        s_A[t]   = 1.f + 2.f * xy + y2;
        s_den[t] = fmaxf(1.f + 2.f * xy + x2s * y2, FMIN);
    }
    __syncthreads();
    for (int i = t; i < 1024; i += 256) {
        int s = i >> 6, c = i & 63;
        s_te[i] = (s_A[s] * (-s_self[c]) + (1.f - x2s) * s_engh[i]) / s_den[s];
    }
    __syncthreads();
    {
        int s = t >> 4, c0 = (t & 15) * 4, base = s * 64 + c0;
        float p = 0; for (int q = 0; q < 4; ++q) { float v = s_te[base + q]; p += v * v; }
        s_part[t] = p;
    }
    __syncthreads();
    if (t < 16) {
        float a = 0; for (int i = 0; i < 16; ++i) a += s_part[t * 16 + i];
        float sn = fmaxf(sqrtf(a), FMIN);
        float lam = fmaxf(2.f / (1.f - x2s), FMIN);
        s_k[t] = (2.f / lam) * artanhf_(sn) / sn;
    }
    __syncthreads();
    if (t < 64) { float a = 0; for (int s = 0; s < 16; ++s) a += s_att[s] * s_k[s] * s_te[s * 64 + t]; s_agg[t] = a; }
    __syncthreads();

    // ---- kg_agg = expmap(agg, self) ----
    s_part[t] = 0.f;
    if (t < 64)       s_part[t] = s_agg[t] * s_agg[t];
    else if (t < 128) s_part[t] = s_self[t - 64] * s_agg[t - 64];
    __syncthreads();
    if (t == 0) { float a = 0, d = 0; for (int i = 0; i < 64; ++i) { a += s_part[i]; d += s_part[64 + i]; } s_scal[2] = a; s_scal[3] = d; }
    __syncthreads();
    {
        float un  = fmaxf(sqrtf(s_scal[2]), FMIN);
        float lam = fmaxf(2.f / (1.f - x2s), FMIN);
        float ce  = tanhf(lam * un * 0.5f) / un;
        float y2 = ce * ce * s_scal[2], xy = ce * s_scal[3];
        float A2 = 1.f + 2.f * xy + y2, B2 = 1.f - x2s;
        float dn = fmaxf(1.f + 2.f * xy + x2s * y2, FMIN);
        if (t < 64) s_kg[t] = (A2 * s_self[t] + B2 * ce * s_agg[t]) / dn;
    }
    __syncthreads();

    // ================= user branch =================
    if (t < 64) { float a = 0; for (int d = 0; d < 64; ++d) a += s_titem[d] * s_Wui[t * 64 + d]; s_imW[t] = a; }
    __syncthreads();
    {
        int s = t >> 4, c0 = (t & 15) * 4, base = s * 64 + c0;
        float p = 0; for (int q = 0; q < 4; ++q) p += s_imW[c0 + q] * s_tu[base + q];
        s_part[t] = p;
    }
    __syncthreads();
    if (t < 16) { float a = 0; for (int i = 0; i < 16; ++i) a += s_part[t * 16 + i]; s_e[t] = a; }
    __syncthreads();
    if (t == 0) {
        float m = -1e30f; for (int s = 0; s < 16; ++s) m = fmaxf(m, s_e[s]);
        float sum = 0.f, ex[16];
        for (int s = 0; s < 16; ++s) { ex[s] = expf(s_e[s] - m); sum += ex[s]; }
        for (int s = 0; s < 16; ++s) s_att[s] = ex[s] / sum;
    }
    __syncthreads();
    if (t < 16) {
        float xy = -s_dotu[t], y2 = s_ns2u[t];
        s_A[t]   = 1.f + 2.f * xy + y2;
        s_den[t] = fmaxf(1.f + 2.f * xy + x2i * y2, FMIN);
    }
    __syncthreads();
    for (int i = t; i < 1024; i += 256) {
        int s = i >> 6, c = i & 63;
        s_tu[i] = (s_A[s] * (-s_item[c]) + (1.f - x2i) * s_ungh[i]) / s_den[s];
    }
    __syncthreads();
    {
        int s = t >> 4, c0 = (t & 15) * 4, base = s * 64 + c0;
        float p = 0; for (int q = 0; q < 4; ++q) { float v = s_tu[base + q]; p += v * v; }
        s_part[t] = p;
    }
    __syncthreads();
    if (t < 16) {
        float a = 0; for (int i = 0; i < 16; ++i) a += s_part[t * 16 + i];
        float sn = fmaxf(sqrtf(a), FMIN);
        float lam = fmaxf(2.f / (1.f - x2i), FMIN);
        s_k[t] = (2.f / lam) * artanhf_(sn) / sn;
    }
    __syncthreads();
    if (t < 64) { float a = 0; for (int s = 0; s < 16; ++s) a += s_att[s] * s_k[s] * s_tu[s * 64 + t]; s_agg[t] = a; }
    __syncthreads();
    s_part[t] = 0.f;
    if (t < 64)       s_part[t] = s_agg[t] * s_agg[t];
    else if (t < 128) s_part[t] = s_item[t - 64] * s_agg[t - 64];
    __syncthreads();
    if (t == 0) { float a = 0, d = 0; for (int i = 0; i < 64; ++i) { a += s_part[i]; d += s_part[64 + i]; } s_scal[4] = a; s_scal[5] = d; }
    __syncthreads();
    {
        float un  = fmaxf(sqrtf(s_scal[4]), FMIN);
        float lam = fmaxf(2.f / (1.f - x2i), FMIN);
        float ce  = tanhf(lam * un * 0.5f) / un;
        float y2 = ce * ce * s_scal[4], xy = ce * s_scal[5];
        float A2 = 1.f + 2.f * xy + y2, B2 = 1.f - x2i;
        float dn = fmaxf(1.f + 2.f * xy + x2i * y2, FMIN);
        if (t < 64) s_ua[t] = (A2 * s_item[t] + B2 * ce * s_agg[t]) / dn;
    }
    __syncthreads();

    // ---- out1 = mobius_add(mobius_add(self, kg), ua) ----
    s_part[t] = 0.f;
    if (t < 64)       s_part[t] = s_kg[t] * s_kg[t];
    else if (t < 128) s_part[t] = s_self[t - 64] * s_kg[t - 64];
    __syncthreads();
    if (t == 0) { float a = 0, d = 0; for (int i = 0; i < 64; ++i) { a += s_part[i]; d += s_part[64 + i]; } s_scal[6] = a; s_scal[7] = d; }
    __syncthreads();
    {
        float y2 = s_scal[6], xy = s_scal[7];
        float A2 = 1.f + 2.f * xy + y2, B2 = 1.f - x2s;
        float dn = fmaxf(1.f + 2.f * xy + x2s * y2, FMIN);
        if (t < 64) s_m1[t] = (A2 * s_self[t] + B2 * s_kg[t]) / dn;
    }
    __syncthreads();
    s_part[t] = 0.f;
    if (t < 64)       s_part[t] = s_m1[t] * s_m1[t];
    else if (t < 128) s_part[t] = s_ua[t - 64] * s_ua[t - 64];
    else if (t < 192) s_part[t] = s_m1[t - 128] * s_ua[t - 128];
    __syncthreads();
    if (t == 0) {
        float a = 0, c2 = 0, d = 0;
        for (int i = 0; i < 64; ++i) { a += s_part[i]; c2 += s_part[64 + i]; d += s_part[128 + i]; }
        s_scal[8] = a; s_scal[9] = c2; s_scal[10] = d;
    }
    __syncthreads();
    {
        float x2 = s_scal[8], y2 = s_scal[9], xy = s_scal[10];
        float A2 = 1.f + 2.f * xy + y2, B2 = 1.f - x2;
        float dn = fmaxf(1.f + 2.f * xy + x2 * y2, FMIN);
        if (t < 64) out1[(size_t)b * 64 + t] = (A2 * s_m1[t] + B2 * s_ua[t]) / dn;
    }
}

// ---------------------------------------------------------------------------
// Kernel B: hyp_linear + final activation.  128 threads (4 waves), 64 rows per
// block.  mx = X @ lin_W^T done with V_WMMA_F32_16X16X4_F32 (fp32-exact path);
// one 16-row tile per wave, lin_W staged in LDS.  Then per-row hyperbolic tail.
// ---------------------------------------------------------------------------
__global__ void __launch_bounds__(128) hyp_linear_kernel(
    const float* __restrict__ X,     // [R,64] (out1 from kernel A)
    const float* __restrict__ linW,  // [64,64]
    const float* __restrict__ linb,  // [64]
    float* __restrict__ out)         // [R,64]
{
    const int t = threadIdx.x;
    const int wave = t >> 5, lane = t & 31;

    __shared__ float s_W[4096];
    __shared__ float s_mx[64 * 64];
    __shared__ float s_hb[64];
    __shared__ float s_red[64];
    __shared__ float s_scal[4];

    for (int i = t; i < 4096; i += 128) s_W[i] = linW[i];
    if (t < 64) s_red[t] = linb[t] * linb[t];
    __syncthreads();
    if (t == 0) {  // hb = proj(expmap0(lin_b)) : coefficient + |hb|^2
        float a = 0; for (int i = 0; i < 64; ++i) a += s_red[i];
        float n  = fmaxf(sqrtf(a), FMIN);
        float th = tanhf(n);
        float hn = fmaxf(th, FMIN);
        float sc = (hn > MAXN) ? (MAXN / hn) : 1.f;
        s_scal[0] = sc * th / n;            // elementwise hb coefficient
        s_scal[1] = (sc * th) * (sc * th);  // |hb|^2
    }
    __syncthreads();
    if (t < 64) s_hb[t] = s_scal[0] * linb[t];

    // ---- WMMA: 16-row tile per wave ----
    const int row0 = blockIdx.x * 64 + wave * 16;
    v8f acc[4];
#pragma unroll
    for (int ct = 0; ct < 4; ++ct)
#pragma unroll
        for (int v = 0; v < 8; ++v) acc[ct][v] = 0.f;

    const int rsel = lane & 15;
    const int koff = (lane >> 4) * 2;
    for (int kc = 0; kc < 16; ++kc) {
        const int k0 = kc * 4;
        const float* ap = X + (size_t)(row0 + rsel) * 64 + k0 + koff;
        v2f a; a.x = ap[0]; a.y = ap[1];    // A[M=rsel, K=koff..koff+1]
#pragma unroll
        for (int ct = 0; ct < 4; ++ct) {
            const float* bp = &s_W[(ct * 16 + rsel) * 64 + k0 + koff];
            v2f bb; bb.x = bp[0]; bb.y = bp[1];  // B[K, N=rsel] = linW[N, K]
            acc[ct] = __builtin_amdgcn_wmma_f32_16x16x4_f32(
                false, a, false, bb, (short)0, acc[ct], false, false);
        }
    }
    // D layout: VGPR v -> M = v + 8*(lane>=16), N = lane%16
#pragma unroll
    for (int ct = 0; ct < 4; ++ct)
#pragma unroll
        for (int v = 0; v < 8; ++v) {
            int M = v + (lane >> 4) * 8;
            int N = ct * 16 + (lane & 15);
            s_mx[(wave * 16 + M) * 64 + N] = acc[ct][v];
        }
    __syncthreads();

    // ---- per-row hyperbolic tail ----
    if (t < 64) {
        const int lr = t;
        const size_t grow = (size_t)blockIdx.x * 64 + t;
        const float* xr = X + grow * 64;
        float xn2 = 0.f;  for (int c = 0; c < 64; ++c) { float v = xr[c]; xn2 += v * v; }
        float mxn2 = 0.f; for (int c = 0; c < 64; ++c) { float v = s_mx[lr * 64 + c]; mxn2 += v * v; }
        float xn  = fmaxf(sqrtf(xn2), FMIN);
        float mxn = fmaxf(sqrtf(mxn2), FMIN);
        // mobius_matvec coefficient (sc=1); mx==0 rows give 0 naturally
        float coef = tanhf(mxn / xn * artanhf_(xn)) / mxn;
        // proj(res)
        float rn = fmaxf(coef * sqrtf(mxn2), FMIN);
        float tot = coef * ((rn > MAXN) ? (MAXN / rn) : 1.f);
        float x2 = tot * tot * mxn2;
        // mobius_add(res, hb)
        float dp = 0.f; for (int c = 0; c < 64; ++c) dp += s_mx[lr * 64 + c] * s_hb[c];
        float xy = tot * dp, y2 = s_scal[1];
        float A2 = 1.f + 2.f * xy + y2, B2 = 1.f - x2;
        float invdn = 1.f / fmaxf(1.f + 2.f * xy + x2 * y2, FMIN);
        float on2 = 0.f;
        for (int c = 0; c < 64; ++c) {
            float v = (A2 * tot * s_mx[lr * 64 + c] + B2 * s_hb[c]) * invdn;
            s_mx[lr * 64 + c] = v; on2 += v * v;
        }
        // proj -> logmap0 -> tanh -> expmap0
        float on  = fmaxf(sqrtf(on2), FMIN);
        float ps2 = (on > MAXN) ? (MAXN / on) : 1.f;
        float n2  = fmaxf(ps2 * sqrtf(on2), FMIN);
        float cl  = artanhf_(n2) / n2 * ps2;
        float wn2 = 0.f;
        for (int c = 0; c < 64; ++c) {
            float w = tanhf(cl * s_mx[lr * 64 + c]);
            s_mx[lr * 64 + c] = w; wn2 += w * w;
        }
        float wn = fmaxf(sqrtf(wn2), FMIN);
        float ce = tanhf(wn) / wn;
        float* orow = out + grow * 64;
        for (int c = 0; c < 64; ++c) orow[c] = ce * s_mx[lr * 64 + c];
    }
}

// ---------------------------------------------------------------------------
extern "C" void kernel_launch(void* const* d_in, const int* in_sizes, int n_in,
                              void* d_out, int out_size, void* d_ws, size_t ws_size,
                              hipStream_t stream) {
    const float* self_v = (const float*)d_in[0];  // (B,1,64)
    const float* ungh_v = (const float*)d_in[1];  // (B,16,64)
    const float* engh_v = (const float*)d_in[2];  // (B,1,16,64)
    const float* item_v = (const float*)d_in[3];  // (B,64)
    const float* Wr     = (const float*)d_in[4];  // (B,1,16,64,64)
    const float* Wui    = (const float*)d_in[5];  // (64,64)
    const float* linW   = (const float*)d_in[6];  // (64,64)
    const float* linb   = (const float*)d_in[7];  // (64,)
    float* out = (float*)d_out;
    float* ws  = (float*)d_ws;                    // out1: B*64 floats

    const int B = in_sizes[0] / 64;               // 2048
    hyp_agg_kernel<<<B, 256, 0, stream>>>(self_v, ungh_v, engh_v, item_v, Wr, Wui, ws);
    hyp_linear_kernel<<<B / 64, 128, 0, stream>>>(ws, linW, linb, out);
}